// DDL_78915729097281
// MI455X (gfx1250) — compile-verified
//
#include <hip/hip_runtime.h>
#include <math.h>

typedef __attribute__((ext_vector_type(2))) float v2f;
typedef __attribute__((ext_vector_type(8))) float v8f;

#define NN 256
#define DD 512
#define NBINS 1000

// ---------------- workspace layout (float element offsets) ----------------
// Tn   : [0,        131072)
// Sn   : [131072,   262144)
// cosT : [262144,   327680)
// cosS : [327680,   393216)
// hist : [393216,   397216)   order: pos_t[1000], neg_t[1000], pos_s[1000], neg_s[1000]
// E    : [397216,   398240)   order: Epos_t[256], Eneg_t[256], Epos_s[256], Eneg_s[256]
// cid  : [398240,   398496)   (ints)
#define WS_TN   0
#define WS_SN   131072
#define WS_COST 262144
#define WS_COSS 327680
#define WS_HIST 393216
#define WS_E    397216
#define WS_CID  398240

// -------------------------------------------------------------------------
// Kernel 1: class ids via argmax over 16 logits + zero histograms
// -------------------------------------------------------------------------
__global__ void argmax_init_kernel(const float* __restrict__ labels,
                                   int* __restrict__ cid,
                                   float* __restrict__ hist) {
    int t = blockIdx.x * blockDim.x + threadIdx.x;
    if (t < NN) {
        const float* row = labels + t * 16;
        float best = row[0];
        int bi = 0;
        #pragma unroll
        for (int c = 1; c < 16; ++c) {
            float v = row[c];
            if (v > best) { best = v; bi = c; }
        }
        cid[t] = bi;
    }
    for (int i = t; i < 4 * NBINS; i += blockDim.x * gridDim.x) hist[i] = 0.0f;
}

// -------------------------------------------------------------------------
// Kernel 2: row L2 normalization (grid.y: 0 = T, 1 = S)
// -------------------------------------------------------------------------
__global__ void __launch_bounds__(256)
normalize_kernel(const float* __restrict__ T, const float* __restrict__ S,
                 float* __restrict__ Tn, float* __restrict__ Sn) {
    const float* src = (blockIdx.y == 0) ? T : S;
    float* dst = (blockIdx.y == 0) ? Tn : Sn;
    int row = blockIdx.x;
    int t = threadIdx.x;
    const float* r = src + row * DD;
    float x0 = r[t];
    float x1 = r[t + 256];
    float ss = x0 * x0 + x1 * x1;
    #pragma unroll
    for (int off = 16; off > 0; off >>= 1) ss += __shfl_down(ss, off, 32);
    __shared__ float red[8];
    int wid = t >> 5, lid = t & 31;
    if (lid == 0) red[wid] = ss;
    __syncthreads();
    __shared__ float inv_s;
    if (t == 0) {
        float tot = 0.0f;
        #pragma unroll
        for (int w = 0; w < 8; ++w) tot += red[w];
        float nrm = sqrtf(tot);
        nrm = fmaxf(nrm, 1e-12f);
        inv_s = 1.0f / nrm;
    }
    __syncthreads();
    float inv = inv_s;
    dst[row * DD + t] = x0 * inv;
    dst[row * DD + t + 256] = x1 * inv;
}

// -------------------------------------------------------------------------
// Kernel 3: cos = Xn * Xn^T via V_WMMA_F32_16X16X4_F32 (one wave per tile)
// A 16x4 f32 layout: lane l -> M = l%16 ; VGPR0 = K=(l<16?0:2), VGPR1 = K+1
// B 4x16 f32 layout mirrors A (rows striped across lane halves)
// D 16x16 f32 layout: VGPR r, lanes 0-15 -> M=r, lanes 16-31 -> M=r+8, N=l%16
// -------------------------------------------------------------------------
__global__ void __launch_bounds__(256)
cos_wmma_kernel(const float* __restrict__ Tn, const float* __restrict__ Sn,
                float* __restrict__ cosT, float* __restrict__ cosS) {
    const float* X = (blockIdx.y == 0) ? Tn : Sn;
    float* C = (blockIdx.y == 0) ? cosT : cosS;

    int wave = (blockIdx.x * blockDim.x + threadIdx.x) >> 5;   // 0..255
    int lane = threadIdx.x & 31;
    int ti = wave >> 4;
    int tj = wave & 15;

    int m = lane & 15;
    int khalf = (lane >> 4) << 1;                              // 0 or 2
    const float* arow = X + (ti * 16 + m) * DD + khalf;
    const float* brow = X + (tj * 16 + m) * DD + khalf;

    v8f c = {};
    #pragma unroll 8
    for (int k = 0; k < DD; k += 4) {
        v2f a = *(const v2f*)(arow + k);
        v2f b = *(const v2f*)(brow + k);
        c = __builtin_amdgcn_wmma_f32_16x16x4_f32(
                /*neg_a=*/false, a, /*neg_b=*/false, b,
                /*c_mod=*/(short)0, c, /*reuse_a=*/false, /*reuse_b=*/false);
    }

    int col  = tj * 16 + m;
    int row0 = ti * 16 + ((lane >> 4) << 3);
    #pragma unroll
    for (int r = 0; r < 8; ++r)
        C[(row0 + r) * NN + col] = c[r];
}

// -------------------------------------------------------------------------
// Kernel 4: fused Gaussian soft-binning histograms (never materialize delta)
// grid (64, 2): 1024 pairs per block; each thread owns 4 contiguous bins.
// -------------------------------------------------------------------------
__global__ void __launch_bounds__(256)
hist_kernel(const float* __restrict__ cosT, const float* __restrict__ cosS,
            const int* __restrict__ cid, float* __restrict__ hist) {
    const float* cosX = (blockIdx.y == 0) ? cosT : cosS;
    float* hp = hist + blockIdx.y * 2 * NBINS;       // pos histogram
    float* hn = hp + NBINS;                          // neg histogram

    __shared__ float sd[1024];
    __shared__ float spw[1024];
    __shared__ float snw[1024];
    __shared__ int   scid[NN];

    int t = threadIdx.x;
    scid[t] = cid[t];
    __syncthreads();

    int base = blockIdx.x * 1024;
    for (int i = t; i < 1024; i += 256) {
        int p  = base + i;
        int pi = p >> 8;
        int pj = p & 255;
        sd[i] = cosX[p];
        bool same = (scid[pi] == scid[pj]);
        spw[i] = (same && (pi != pj)) ? 1.0f : 0.0f;
        snw[i] = same ? 0.0f : 1.0f;
    }
    __syncthreads();

    int b0 = t * 4;
    if (b0 >= NBINS) return;

    float tc[4];
    #pragma unroll
    for (int r = 0; r < 4; ++r) tc[r] = -1.0f + 0.002f * (float)(b0 + r);
    float tmid = tc[1] + 0.001f;

    float pa[4] = {0.f, 0.f, 0.f, 0.f};
    float na[4] = {0.f, 0.f, 0.f, 0.f};

    for (int i = 0; i < 1024; ++i) {
        float d  = sd[i];
        float pw = spw[i];
        float nw = snw[i];
        if (pw + nw == 0.0f) continue;              // diagonal
        if (fabsf(d - tmid) > 0.72f) continue;      // >~7 sigma: < 2e-11
        #pragma unroll
        for (int r = 0; r < 4; ++r) {
            float z = (d - tc[r]) * 10.0f;          // 1/SIGMA = 10
            float e = __expf(-0.5f * z * z);
            pa[r] = fmaf(pw, e, pa[r]);
            na[r] = fmaf(nw, e, na[r]);
        }
    }
    #pragma unroll
    for (int r = 0; r < 4; ++r) {
        atomicAdd(&hp[b0 + r], pa[r]);
        atomicAdd(&hn[b0 + r], na[r]);
    }
}

// -------------------------------------------------------------------------
// block reduce helper: result valid on thread 0 only
// -------------------------------------------------------------------------
__device__ float block_reduce(float v, float* red) {
    #pragma unroll
    for (int off = 16; off > 0; off >>= 1) v += __shfl_down(v, off, 32);
    int wid = threadIdx.x >> 5, lid = threadIdx.x & 31;
    if (lid == 0) red[wid] = v;
    __syncthreads();
    float r = 0.0f;
    if (threadIdx.x == 0) {
        int nw = blockDim.x >> 5;
        for (int w = 0; w < nw; ++w) r += red[w];
    }
    __syncthreads();
    return r;
}

// -------------------------------------------------------------------------
// Kernel 5: per-row weighted means E_pos / E_neg. grid (256, 2)
// -------------------------------------------------------------------------
__global__ void __launch_bounds__(256)
e_kernel(const float* __restrict__ cosT, const float* __restrict__ cosS,
         const int* __restrict__ cid, float* __restrict__ E) {
    const float* cosX = (blockIdx.y == 0) ? cosT : cosS;
    int row = blockIdx.x;
    int t = threadIdx.x;
    int myc = cid[row];
    float cv = cosX[row * NN + t];
    bool same = (cid[t] == myc);
    float pw = (same && (t != row)) ? 1.0f : 0.0f;
    float nw = same ? 0.0f : 1.0f;

    __shared__ float red[8];
    float psum = block_reduce(pw * cv, red);
    float pcnt = block_reduce(pw, red);
    float nsum = block_reduce(nw * cv, red);
    float ncnt = block_reduce(nw, red);
    if (t == 0) {
        E[blockIdx.y * 512 + row]       = psum / pcnt;
        E[blockIdx.y * 512 + 256 + row] = nsum / ncnt;
    }
}

// -------------------------------------------------------------------------
// Kernel 6: final scalar. 1 block x 256 threads.
// -------------------------------------------------------------------------
__global__ void __launch_bounds__(256)
final_kernel(const int* __restrict__ cid, const float* __restrict__ hist,
             const float* __restrict__ E, float* __restrict__ out) {
    int t = threadIdx.x;
    __shared__ float cnt[16];
    if (t < 16) cnt[t] = 0.0f;
    __syncthreads();
    atomicAdd(&cnt[cid[t]], 1.0f);
    __syncthreads();

    __shared__ float invP_s, invN_s;
    if (t == 0) {
        float ps = 0.0f, sq = 0.0f;
        #pragma unroll
        for (int c = 0; c < 16; ++c) {
            float n = cnt[c];
            ps += n * (n - 1.0f);
            sq += n * n;
        }
        invP_s = 1.0f / ps;
        invN_s = 1.0f / ((float)NN * (float)NN - sq);
    }
    __syncthreads();
    float invP = invP_s, invN = invN_s;

    const float EPS = 1e-9f;
    float pos_kl = 0.0f, neg_kl = 0.0f;
    for (int b = t; b < NBINS; b += 256) {
        float pa = hist[b]          * invP + EPS;   // pos_t
        float pd = hist[2000 + b]   * invP + EPS;   // pos_s
        pos_kl += pa * (logf(pa) - logf(pd));
        float na = hist[1000 + b]   * invN + EPS;   // neg_t
        float nd = hist[3000 + b]   * invN + EPS;   // neg_s
        neg_kl += na * (logf(na) - logf(nd));
    }
    float o1 = fabsf(E[t]       - E[512 + t]);      // |E_pos_t - E_pos_s|
    float o2 = fabsf(E[256 + t] - E[768 + t]);      // |E_neg_t - E_neg_s|
    float o3 = E[512 + t] - E[768 + t];             //  E_pos_s - E_neg_s

    __shared__ float red[8];
    float pk = block_reduce(pos_kl, red);
    float nk = block_reduce(neg_kl, red);
    float s1 = block_reduce(o1, red);
    float s2 = block_reduce(o2, red);
    float s3 = block_reduce(o3, red);
    if (t == 0) {
        float order = 0.5f * ((s1 + s2 + s3) * (1.0f / 256.0f));
        out[0] = 0.1f * pk + 0.02f * nk + order;
    }
}

// -------------------------------------------------------------------------
extern "C" void kernel_launch(void* const* d_in, const int* in_sizes, int n_in,
                              void* d_out, int out_size, void* d_ws, size_t ws_size,
                              hipStream_t stream) {
    const float* T      = (const float*)d_in[0];   // [256,512]
    const float* S      = (const float*)d_in[1];   // [256,512]
    const float* labels = (const float*)d_in[2];   // [256,16]
    float* out = (float*)d_out;

    float* ws   = (float*)d_ws;
    float* Tn   = ws + WS_TN;
    float* Sn   = ws + WS_SN;
    float* cosT = ws + WS_COST;
    float* cosS = ws + WS_COSS;
    float* hist = ws + WS_HIST;
    float* E    = ws + WS_E;
    int*   cid  = (int*)(ws + WS_CID);

    argmax_init_kernel<<<16, 256, 0, stream>>>(labels, cid, hist);
    normalize_kernel<<<dim3(NN, 2), 256, 0, stream>>>(T, S, Tn, Sn);
    cos_wmma_kernel<<<dim3(32, 2), 256, 0, stream>>>(Tn, Sn, cosT, cosS);
    hist_kernel<<<dim3(64, 2), 256, 0, stream>>>(cosT, cosS, cid, hist);
    e_kernel<<<dim3(NN, 2), 256, 0, stream>>>(cosT, cosS, cid, E);
    final_kernel<<<1, 256, 0, stream>>>(cid, hist, E, out);
}